// RelationMultiHeadSelfAttention_74672301408580
// MI455X (gfx1250) — compile-verified
//
#include <hip/hip_runtime.h>
#include <hip/hip_bf16.h>

// ---------------------------------------------------------------------------
// Mathematically the reference collapses:
//   einsum('nhqk,nqhd->nqhd', softmax_k(...), v) == v * sum_k(attn) == v
// so out = concat(cls, values) @ blockdiag(Wv)^T @ W_out^T + b_out.
// relation / keys / query / mask cancel exactly.
// ---------------------------------------------------------------------------

typedef __attribute__((ext_vector_type(2))) float v2f;
typedef __attribute__((ext_vector_type(8))) float v8f;

#define EMBED 512
#define HEAD_DIM 64
#define QLEN 256   // L+1
#define NROWS 1024 // N * QLEN

// Kernel 1: fold per-head Wv into W_out.
// C[o, h*64+d] = sum_e W_out[o, h*64+e] * Wv[e, d]     (C is 512x512 row-major)
__global__ void build_combined_kernel(const float* __restrict__ W_out,
                                      const float* __restrict__ Wv,
                                      float* __restrict__ C) {
    int idx = blockIdx.x * blockDim.x + threadIdx.x;   // 0 .. 512*512-1
    int o = idx >> 9;
    int i = idx & (EMBED - 1);
    int h = i >> 6;
    int d = i & (HEAD_DIM - 1);
    const float* wrow = W_out + (size_t)o * EMBED + h * HEAD_DIM;
    float acc = 0.0f;
#pragma unroll 8
    for (int e = 0; e < HEAD_DIM; ++e) {
        acc = fmaf(wrow[e], Wv[e * HEAD_DIM + d], acc);
    }
    C[idx] = acc;
}

// Kernel 2: out(1024x512) = X(1024x512) @ C^T + b_out, via V_WMMA_F32_16X16X4_F32.
// X rows are gathered: row g -> (g&255)==0 ? cls_token : values[g>>8, (g&255)-1, :].
// One wave (32 lanes) computes one 16x16 output tile; K loop in steps of 4.
__global__ void fused_out_gemm_kernel(const float* __restrict__ values,
                                      const float* __restrict__ cls_token,
                                      const float* __restrict__ C,
                                      const float* __restrict__ b_out,
                                      float* __restrict__ out) {
    const int wave = threadIdx.x >> 5;              // 0..7
    const int lane = threadIdx.x & 31;
    const int tile = blockIdx.x * 8 + wave;         // 0..2047
    const int tiles_n = EMBED / 16;                 // 32 col tiles
    const int row0 = (tile / tiles_n) * 16;         // 0..1008
    const int col0 = (tile % tiles_n) * 16;         // 0..496

    const int m = lane & 15;                        // row/col within tile
    const int h = lane >> 4;                        // half selector (K pairs)

    // A-fragment source: fixed X row for this lane (ISA 7.12.2: A 16x4 f32,
    // lanes 0-15 hold K=0,1 ; lanes 16-31 hold K=2,3 for row M=lane%16).
    const int grow = row0 + m;                      // global X row
    const int img  = grow >> 8;                     // batch index
    const int l    = grow & (QLEN - 1);             // position incl. cls
    const float* xrow = (l == 0)
        ? cls_token
        : values + ((size_t)(img * (QLEN - 1) + (l - 1))) * EMBED;

    // B-fragment source: B[kk, n] = C[col0+n, k+kk]; lane n=m reads row col0+m
    // of C, pair {k+2h, k+2h+1} (mirrors the A K-split across lane halves).
    const float* crow = C + (size_t)(col0 + m) * EMBED;

    v8f acc = {};
#pragma unroll 4
    for (int k = 0; k < EMBED; k += 4) {
        v2f a = *(const v2f*)(xrow + k + 2 * h);
        v2f b = *(const v2f*)(crow + k + 2 * h);
        acc = __builtin_amdgcn_wmma_f32_16x16x4_f32(
            /*neg_a=*/false, a, /*neg_b=*/false, b,
            /*c_mod=*/(short)0, acc, /*reuse_a=*/false, /*reuse_b=*/false);
    }

    // D layout (ISA 7.12.2): VGPR r -> element (M = r + 8h, N = m).
    const float bo = b_out[col0 + m];
    float* obase = out + (size_t)(row0 + 8 * h) * EMBED + (col0 + m);
#pragma unroll
    for (int r = 0; r < 8; ++r) {
        obase[(size_t)r * EMBED] = acc[r] + bo;
    }
}

extern "C" void kernel_launch(void* const* d_in, const int* in_sizes, int n_in,
                              void* d_out, int out_size, void* d_ws, size_t ws_size,
                              hipStream_t stream) {
    // setup_inputs() order:
    // 0 values, 1 keys, 2 query, 3 mask, 4 relation, 5 cls_token,
    // 6 Wv, 7 Wk, 8 Wq, 9 W_out, 10 b_out
    const float* values    = (const float*)d_in[0];
    const float* cls_token = (const float*)d_in[5];
    const float* Wv        = (const float*)d_in[6];
    const float* W_out     = (const float*)d_in[9];
    const float* b_out     = (const float*)d_in[10];
    float* out = (float*)d_out;
    float* C   = (float*)d_ws;   // 512*512*4 = 1 MB scratch

    (void)in_sizes; (void)n_in; (void)out_size; (void)ws_size;

    // 512*512 elements / 256 threads = 1024 blocks
    build_combined_kernel<<<1024, 256, 0, stream>>>(W_out, Wv, C);

    // 2048 tiles, 8 waves (256 threads) per block -> 256 blocks
    fused_out_gemm_kernel<<<256, 256, 0, stream>>>(values, cls_token, C, b_out, out);
}